// MomentAccumulatorObserver_13786845020652
// MI455X (gfx1250) — compile-verified
//
#include <hip/hip_runtime.h>
#include <stdint.h>
#include <stddef.h>

// ---------------------------------------------------------------------------
// MomentAccumulatorObserver: scatter + gather/product accumulate.
// Pure memory-bound (≈0.1 FLOP/byte): optimize data movement, not math.
//  - flat_state (4MB) stays hot in MI455X's 192MB L2 -> random gathers cheap.
//  - streaming operands (edges 64MB, carry2 32MB) staged via CDNA5 async
//    global->LDS data mover with double buffering (ASYNCcnt pipeline).
// ---------------------------------------------------------------------------

#ifndef __has_builtin
#define __has_builtin(x) 0
#endif

#if __has_builtin(__builtin_amdgcn_global_load_async_to_lds_b128) && \
    __has_builtin(__builtin_amdgcn_s_wait_asynccnt)
#define HAVE_ASYNC_LDS 1
#else
#define HAVE_ASYNC_LDS 0
#endif

#define BLOCK 256
#define EPT 4                    // edges per thread
#define TILE (BLOCK * EPT)       // 1024 edges per tile

// True clang vector types (HIP_vector_type structs are rejected by the
// nontemporal builtins and by the async-LDS builtin signatures).
typedef int   v4i __attribute__((ext_vector_type(4)));
typedef float v4f __attribute__((ext_vector_type(4)));

#if HAVE_ASYNC_LDS
typedef __attribute__((address_space(1))) v4i* gas_v4i_t;
typedef __attribute__((address_space(3))) v4i* lds_v4i_t;
__device__ __forceinline__ void async_copy_b128(const void* g, void* l) {
  // 16B per lane, global -> LDS, tracked by ASYNCcnt
  __builtin_amdgcn_global_load_async_to_lds_b128((gas_v4i_t)g, (lds_v4i_t)l,
                                                 /*offset=*/0, /*cpol=*/0);
}
#endif

// ---- pass 1: flat_state[idx[i]] = vals[i] (idx is a permutation) ----------
__global__ void k_scatter(const float* __restrict__ vals,
                          const int* __restrict__ idx,
                          float* __restrict__ flat, int n) {
  int i = (blockIdx.x * BLOCK + threadIdx.x) * 4;
  if (i + 4 <= n) {
    v4i p = __builtin_nontemporal_load((const v4i*)(idx + i));
    v4f v = __builtin_nontemporal_load((const v4f*)(vals + i));
    flat[p.x] = v.x;
    flat[p.y] = v.y;
    flat[p.z] = v.z;
    flat[p.w] = v.w;
  } else {
    for (int k = 0; k < 4; ++k) {
      int j = i + k;
      if (j < n) flat[idx[j]] = vals[j];
    }
  }
}

// ---- pass 2: mem1[i] = carry1[i] + flat[m1[i]] ----------------------------
__global__ void k_mem1(const int* __restrict__ m1,
                       const float* __restrict__ c1,
                       const float* __restrict__ flat,
                       float* __restrict__ out1, int n) {
  int i = (blockIdx.x * BLOCK + threadIdx.x) * 4;
  if (i + 4 <= n) {
    v4i p = __builtin_nontemporal_load((const v4i*)(m1 + i));
    v4f c = __builtin_nontemporal_load((const v4f*)(c1 + i));
    v4f r;
    r.x = c.x + flat[p.x];
    r.y = c.y + flat[p.y];
    r.z = c.z + flat[p.z];
    r.w = c.w + flat[p.w];
    __builtin_nontemporal_store(r, (v4f*)(out1 + i));
  } else {
    for (int k = 0; k < 4; ++k) {
      int j = i + k;
      if (j < n) out1[j] = c1[j] + flat[m1[j]];
    }
  }
}

// ---- pass 3: mem2[e] = carry2[e] + flat[edge0]*flat[edge1] ----------------
// Streaming edge indices + carry are double-buffered through LDS with the
// CDNA5 async data mover; the flat[] gathers hit L2 (table is 4MB).
__global__ void k_mem2(const int* __restrict__ edges,   // 2*n ints, interleaved
                       const float* __restrict__ carry2,
                       const float* __restrict__ flat,
                       float* __restrict__ out2, unsigned n) {
  const unsigned t = threadIdx.x;
  const unsigned tiles = (n + TILE - 1) / TILE;

#if HAVE_ASYNC_LDS
  __shared__ int s_e[2][TILE * 2];   // 2 x 8KB edge-index staging
  __shared__ float s_c[2][TILE];     // 2 x 4KB carry staging

  unsigned tile = blockIdx.x;
  int buf = 0;
  bool cur_staged = false;
  if (tile < tiles && (tile + 1) * (unsigned)TILE <= n) {
    const int* ge = edges + (size_t)tile * (TILE * 2) + t * 8;
    async_copy_b128(ge, &s_e[0][t * 8]);
    async_copy_b128(ge + 4, &s_e[0][t * 8 + 4]);
    async_copy_b128(carry2 + (size_t)tile * TILE + t * 4, &s_c[0][t * 4]);
    cur_staged = true;
  }
  while (tile < tiles) {
    const unsigned next = tile + gridDim.x;
    const bool next_full = (next < tiles) && ((next + 1) * (unsigned)TILE <= n);
    if (next_full) {  // prefetch next tile into the other buffer
      const int nb = buf ^ 1;
      const int* ge = edges + (size_t)next * (TILE * 2) + t * 8;
      async_copy_b128(ge, &s_e[nb][t * 8]);
      async_copy_b128(ge + 4, &s_e[nb][t * 8 + 4]);
      async_copy_b128(carry2 + (size_t)next * TILE + t * 4, &s_c[nb][t * 4]);
    }
    if (cur_staged) {
      // Async loads complete in order: <=3 outstanding means the current
      // tile's 3 ops (issued before the stage above) have landed in LDS.
      if (next_full) __builtin_amdgcn_s_wait_asynccnt(3);
      else           __builtin_amdgcn_s_wait_asynccnt(0);
      const unsigned base = tile * TILE;
      const int a0 = s_e[buf][t * 8 + 0], b0 = s_e[buf][t * 8 + 1];
      const int a1 = s_e[buf][t * 8 + 2], b1 = s_e[buf][t * 8 + 3];
      const int a2 = s_e[buf][t * 8 + 4], b2 = s_e[buf][t * 8 + 5];
      const int a3 = s_e[buf][t * 8 + 6], b3 = s_e[buf][t * 8 + 7];
      v4f r;
      r.x = s_c[buf][t * 4 + 0] + flat[a0] * flat[b0];
      r.y = s_c[buf][t * 4 + 1] + flat[a1] * flat[b1];
      r.z = s_c[buf][t * 4 + 2] + flat[a2] * flat[b2];
      r.w = s_c[buf][t * 4 + 3] + flat[a3] * flat[b3];
      __builtin_nontemporal_store(r, (v4f*)(out2 + base + t * 4));
    } else {
      // partial (last) tile: direct scalar path
      const unsigned base = tile * TILE;
      for (int k = 0; k < EPT; ++k) {
        unsigned e = base + t * EPT + k;
        if (e < n) {
          int a = edges[2 * (size_t)e];
          int b = edges[2 * (size_t)e + 1];
          out2[e] = carry2[e] + flat[a] * flat[b];
        }
      }
    }
    cur_staged = next_full;
    buf ^= 1;
    tile = next;
  }
#else
  // Fallback: vectorized direct loads, grid-stride over tiles.
  for (unsigned tile = blockIdx.x; tile < tiles; tile += gridDim.x) {
    const unsigned e0 = tile * TILE + t * EPT;
    if (e0 + EPT <= n) {
      v4i p0 = __builtin_nontemporal_load((const v4i*)(edges + (size_t)e0 * 2));
      v4i p1 = __builtin_nontemporal_load((const v4i*)(edges + (size_t)e0 * 2 + 4));
      v4f c = __builtin_nontemporal_load((const v4f*)(carry2 + e0));
      v4f r;
      r.x = c.x + flat[p0.x] * flat[p0.y];
      r.y = c.y + flat[p0.z] * flat[p0.w];
      r.z = c.z + flat[p1.x] * flat[p1.y];
      r.w = c.w + flat[p1.z] * flat[p1.w];
      __builtin_nontemporal_store(r, (v4f*)(out2 + e0));
    } else {
      for (int k = 0; k < EPT; ++k) {
        unsigned e = e0 + k;
        if (e < n) {
          int a = edges[2 * (size_t)e];
          int b = edges[2 * (size_t)e + 1];
          out2[e] = carry2[e] + flat[a] * flat[b];
        }
      }
    }
  }
#endif
}

extern "C" void kernel_launch(void* const* d_in, const int* in_sizes, int n_in,
                              void* d_out, int out_size, void* d_ws, size_t ws_size,
                              hipStream_t stream) {
  const float* vals = (const float*)d_in[0];   // sampled_values [N_NODES]
  const int* sidx   = (const int*)d_in[1];     // scatter_index  [N_NODES]
  const int* m1     = (const int*)d_in[2];     // moment_slices_1 [N_NODES,1]
  const int* m2     = (const int*)d_in[3];     // moment_slices_2 [N_EDGES,2]
  const float* c1   = (const float*)d_in[4];   // carry1 [N_NODES]
  const float* c2   = (const float*)d_in[5];   // carry2 [N_EDGES]

  const int n_nodes = in_sizes[0];
  const int n_edges = in_sizes[5];

  float* flat = (float*)d_ws;                  // 4MB scratch: flat_state
  float* out1 = (float*)d_out;                 // mem1 [n_nodes]
  float* out2 = out1 + n_nodes;                // mem2 [n_edges]

  const int nodes_blocks = (n_nodes + BLOCK * 4 - 1) / (BLOCK * 4);
  k_scatter<<<nodes_blocks, BLOCK, 0, stream>>>(vals, sidx, flat, n_nodes);
  k_mem1<<<nodes_blocks, BLOCK, 0, stream>>>(m1, c1, flat, out1, n_nodes);

  const unsigned tiles = ((unsigned)n_edges + TILE - 1) / TILE;
  unsigned grid = tiles < 2048u ? tiles : 2048u;  // ~4 tiles/block -> pipeline
  if (grid == 0) grid = 1;
  k_mem2<<<grid, BLOCK, 0, stream>>>(m2, c2, flat, out2, (unsigned)n_edges);
}